// ParallelExperts_50878182588545
// MI455X (gfx1250) — compile-verified
//
#include <hip/hip_runtime.h>

typedef __attribute__((ext_vector_type(2))) float v2f;
typedef __attribute__((ext_vector_type(8))) float v8f;

#define D_MODEL        128
#define ROWS_PER_BLOCK 128
#define LDS_STRIDE     132   // 128 + 4; stride % 64 == 4 -> conflict-free b64 fragment loads
#define THREADS        256   // 8 wave32s

// Dynamic LDS layout (floats):
//   lds_x : ROWS_PER_BLOCK * LDS_STRIDE   (gathered input rows, padded stride)
//   lds_w : 128 * LDS_STRIDE              (current expert weight [d_out][d_in], padded stride)
//   lds_g : 128                           (per-row gate)
//   lds_t : 128 ints                      (per-row token id)
//   lds_e : 128 ints                      (per-row expert id)

__global__ __launch_bounds__(THREADS, 1)
void moe_scatter2scatter_wmma(const float* __restrict__ inputs,
                              const float* __restrict__ weight,
                              const float* __restrict__ gates,
                              const int*   __restrict__ sorted_expert_idxs,
                              const int*   __restrict__ sorted_scattered_idxs,
                              float*       __restrict__ out)
{
    extern __shared__ float smem[];
    float* lds_x = smem;                                    // 128*132 floats
    float* lds_w = lds_x + ROWS_PER_BLOCK * LDS_STRIDE;     // 128*132 floats
    float* lds_g = lds_w + 128 * LDS_STRIDE;                // 128 floats
    int*   lds_t = (int*)(lds_g + 128);                     // 128 ints
    int*   lds_e = lds_t + 128;                             // 128 ints

    const int tid  = threadIdx.x;
    const int base = blockIdx.x * ROWS_PER_BLOCK;

    // ---- Phase 0: per-row metadata (expert, token, gate) ----
    if (tid < ROWS_PER_BLOCK) {
        int s = base + tid;
        int p = sorted_scattered_idxs[s];        // original flat slot index
        lds_e[tid] = sorted_expert_idxs[s];
        lds_t[tid] = p >> 1;                     // TOP_K == 2
        lds_g[tid] = gates[p];                   // gates viewed flat [N*k]
    }
    __syncthreads();

    // ---- Phase 1: gather input rows into LDS (coalesced 16B segments) ----
    for (int idx = tid; idx < ROWS_PER_BLOCK * 32; idx += THREADS) {
        int row = idx >> 5, seg = idx & 31;
        float4 v = *(const float4*)(inputs + (size_t)lds_t[row] * D_MODEL + seg * 4);
        *(float4*)(lds_x + row * LDS_STRIDE + seg * 4) = v;
    }

    const int wave  = tid >> 5;
    const int lane  = tid & 31;
    const int mrow  = lane & 15;               // A-row / B-col within a 16x16 tile
    const int khalf = (lane >> 4) << 1;        // 0 or 2: K sub-pair this lane holds
    const int rowA  = (wave << 4) + mrow;      // block-local slot row of this lane's A frag

    v8f acc[8];
    #pragma unroll
    for (int n = 0; n < 8; ++n) acc[n] = {};

    __syncthreads();
    const int eFirst = lds_e[0];
    const int eLast  = lds_e[ROWS_PER_BLOCK - 1];  // sorted -> contiguous expert range

    for (int e = eFirst; e <= eLast; ++e) {
        __syncthreads();   // previous iteration done reading lds_w
        // Stage weight[e] ([d_out][d_in] row-major) into padded LDS, coalesced.
        const float* wsrc = weight + (size_t)e * D_MODEL * D_MODEL;
        for (int idx = tid; idx < 128 * 32; idx += THREADS) {
            int row = idx >> 5, seg = idx & 31;
            *(float4*)(lds_w + row * LDS_STRIDE + seg * 4) =
                *(const float4*)(wsrc + row * D_MODEL + seg * 4);
        }
        __syncthreads();

        // Rows whose expert != e contribute zero (data mask; EXEC stays all-1s).
        const float  mask  = (lds_e[rowA] == e) ? 1.0f : 0.0f;
        const float* aBase = lds_x + rowA * LDS_STRIDE + khalf;
        const float* bBase = lds_w + mrow * LDS_STRIDE + khalf;

        #pragma unroll 4
        for (int kk = 0; kk < 32; ++kk) {          // K = 128 in steps of 4
            v2f a = *(const v2f*)(aBase + kk * 4); // A[mrow][4kk+khalf .. +1]
            a *= mask;
            #pragma unroll
            for (int n = 0; n < 8; ++n) {          // 8 column tiles = 128 outputs
                // B[k][col] = weight[e][col][k]; col = n*16 + mrow
                v2f b = *(const v2f*)(bBase + n * 16 * LDS_STRIDE + kk * 4);
                acc[n] = __builtin_amdgcn_wmma_f32_16x16x4_f32(
                    false, a, false, b, (short)0, acc[n], false, false);
            }
        }
    }

    // ---- Epilogue: gated atomic scatter-add into out[token][col] ----
    __syncthreads();
    #pragma unroll
    for (int v = 0; v < 8; ++v) {
        int   rowC = (wave << 4) + v + ((lane >> 4) << 3);  // C layout: M = v + 8*(lane/16)
        float g    = lds_g[rowC];
        int   tok  = lds_t[rowC];
        float* dst = out + (size_t)tok * D_MODEL + mrow;
        #pragma unroll
        for (int n = 0; n < 8; ++n) {
            atomicAdd(dst + n * 16, g * acc[n][v]);
        }
    }
}

extern "C" void kernel_launch(void* const* d_in, const int* in_sizes, int n_in,
                              void* d_out, int out_size, void* d_ws, size_t ws_size,
                              hipStream_t stream) {
    const float* inputs = (const float*)d_in[0];   // [N, 128] f32
    const float* weight = (const float*)d_in[1];   // [8, 128, 128] f32
    const float* gates  = (const float*)d_in[2];   // [N, 2] f32
    const int*   sei    = (const int*)d_in[4];     // [N*k] sorted expert idxs
    const int*   ssi    = (const int*)d_in[5];     // [N*k] sorted scattered idxs
    float*       out    = (float*)d_out;           // [N, 128] f32

    const size_t smem = (size_t)(2 * ROWS_PER_BLOCK * LDS_STRIDE + 128) * sizeof(float)
                      + (size_t)2 * 128 * sizeof(int);   // ~136.7 KB (< 320 KB/WGP)
    hipFuncSetAttribute((const void*)moe_scatter2scatter_wmma,
                        hipFuncAttributeMaxDynamicSharedMemorySize, (int)smem);

    // Atomic accumulation target must start at zero (d_out is poisoned).
    hipMemsetAsync(d_out, 0, (size_t)out_size * sizeof(float), stream);

    const int S = in_sizes[4];                 // N*k = 524288, multiple of 128
    dim3 grid(S / ROWS_PER_BLOCK);             // 4096 workgroups
    moe_scatter2scatter_wmma<<<grid, THREADS, smem, stream>>>(
        inputs, weight, gates, sei, ssi, out);
}